// DeformRoIPooling_62886911148489
// MI455X (gfx1250) — compile-verified
//
#include <hip/hip_runtime.h>

// Deformable PSRoI pooling for MI455X (gfx1250, wave32).
// Gather-bound: feature map (125 MB) fits in 192 MB L2; HBM floor ~6us.
// Bilinear blend + 16-sample reduction via v_wmma_f32_16x16x4_f32 in full f32:
// A = corner weights (all M rows identical), B = gathered corner values for 16
// channels, accumulation chained through the C operand.
// Round-2 polish: int4 sample records (1x ds_load_b128), per-bin SALU validity
// bitmask, scalar sample offset + 32-bit lane voffset (saddr-form gathers).

typedef float v2f __attribute__((ext_vector_type(2)));
typedef float v8f __attribute__((ext_vector_type(8)));

#define SCALE      0.0625f
#define POOLP      7
#define C_FEAT     256
#define H_FEAT     200
#define W_FEAT     304
#define TRANS_STD  0.1f
#define NBIN       (POOLP * POOLP)     // 49
#define NSAMP      (NBIN * 16)         // 784
#define CH_PER_BLK 128
#define BLK_THREADS 128
#define PLANE      (H_FEAT * W_FEAT)   // 60800 elements per channel plane

__global__ __launch_bounds__(BLK_THREADS)
void deform_roi_pool_wmma(const float* __restrict__ data,
                          const float* __restrict__ rois,
                          const float* __restrict__ offset,
                          float* __restrict__ out)
{
    // one record per sample: {bitcast(dx), bitcast(dy), elem_off(y0*W+x0), valid}
    __shared__ int4     s_rec[NSAMP];
    __shared__ unsigned s_mask[NBIN];  // 16-bit validity mask per bin
    __shared__ float    s_inv[NBIN];   // 1/cnt (0 if cnt==0)
    __shared__ float    s_res[CH_PER_BLK * NBIN];

    const int r   = blockIdx.x;        // roi index
    const int cg  = blockIdx.y;        // channel slab (0/1)
    const int tid = threadIdx.x;

    // ---- per-ROI geometry (uniform) ----
    const float* roi = rois + r * 5;
    const int   bs  = __builtin_amdgcn_readfirstlane((int)roi[0]);
    const float rsw = rintf(roi[1]) * SCALE - 0.5f;            // jnp.round == rintf
    const float rsh = rintf(roi[2]) * SCALE - 0.5f;
    const float rew = (rintf(roi[3]) + 1.0f) * SCALE - 0.5f;
    const float reh = (rintf(roi[4]) + 1.0f) * SCALE - 0.5f;
    const float roi_w = fmaxf(rew - rsw, 0.1f);
    const float roi_h = fmaxf(reh - rsh, 0.1f);
    const float bin_w = roi_w * (1.0f / POOLP);
    const float bin_h = roi_h * (1.0f / POOLP);
    const float sub_w = bin_w * 0.25f;
    const float sub_h = bin_h * 0.25f;

    // ---- phase 1: build 49x16 sample table (channel-independent) ----
    for (int id = tid; id < NSAMP; id += BLK_THREADS) {
        const int bin = id >> 4;
        const int s   = id & 15;
        const int ph  = bin / POOLP;
        const int pw  = bin % POOLP;
        const int ih  = s >> 2;
        const int iw  = s & 3;

        const float tx = offset[((r * 2 + 0) * POOLP + ph) * POOLP + pw] * TRANS_STD;
        const float ty = offset[((r * 2 + 1) * POOLP + ph) * POOLP + pw] * TRANS_STD;

        const float wpos = (float)pw * bin_w + rsw + tx * roi_w + (float)iw * sub_w;
        const float hpos = (float)ph * bin_h + rsh + ty * roi_h + (float)ih * sub_h;

        const int valid = (wpos > -0.5f) & (wpos < (float)W_FEAT - 0.5f) &
                          (hpos > -0.5f) & (hpos < (float)H_FEAT - 0.5f);

        const float wc = fminf(fmaxf(wpos, 0.0f), (float)(W_FEAT - 1));
        const float hc = fminf(fmaxf(hpos, 0.0f), (float)(H_FEAT - 1));
        // Rebase so (x0,x0+1),(y0,y0+1) are always in-bounds; exact equivalent
        // of the reference's x1=min(x0+1,W-1) clamp.
        const int x0 = min((int)floorf(wc), W_FEAT - 2);
        const int y0 = min((int)floorf(hc), H_FEAT - 2);

        int4 rec;
        rec.x = __float_as_int(wc - (float)x0);   // dx
        rec.y = __float_as_int(hc - (float)y0);   // dy
        rec.z = y0 * W_FEAT + x0;                 // element offset in plane
        rec.w = valid;
        s_rec[id] = rec;
    }
    __syncthreads();

    if (tid < NBIN) {
        unsigned m = 0;
        int cnt = 0;
        for (int s = 0; s < 16; ++s) {
            const int v = s_rec[tid * 16 + s].w;
            m |= ((unsigned)v) << s;
            cnt += v;
        }
        s_mask[tid] = m;
        s_inv[tid]  = (cnt > 0) ? (1.0f / (float)cnt) : 0.0f;
    }
    __syncthreads();

    // ---- phase 2: WMMA accumulation, 32 channels per wave ----
    const int wave = tid >> 5;
    const int lane = tid & 31;
    const int half = lane >> 4;        // 0 -> K0/K1 rows (y0), 1 -> K2/K3 (y0+1)
    const int cA   = cg * CH_PER_BLK + wave * 32 + (lane & 15);  // chain-A channel

    // uniform batch base pointer; per-lane 32-bit element displacement
    const float* dataB = data + (size_t)bs * C_FEAT * PLANE;
    const int vIdxA = cA * PLANE + half * W_FEAT;                // < 2^24 elems
    const int vIdxB = vIdxA + 16 * PLANE;                        // chain-B channel

    const float hbase = half ? 0.0f : 1.0f;   // e = half ? dy : (1-dy)
    const float hsel  = half ? 1.0f : -1.0f;  //   = hbase + hsel*dy

    for (int bin = 0; bin < NBIN; ++bin) {
        const unsigned m =
            (unsigned)__builtin_amdgcn_readfirstlane((int)s_mask[bin]);
        v8f accA = {0.f, 0.f, 0.f, 0.f, 0.f, 0.f, 0.f, 0.f};
        v8f accB = {0.f, 0.f, 0.f, 0.f, 0.f, 0.f, 0.f, 0.f};
        #pragma unroll
        for (int s = 0; s < 16; ++s) {
            if (m & (1u << s)) {      // SGPR bit-test -> scalar branch, EXEC full
                const int4 rec = s_rec[bin * 16 + s];
                const int soff = __builtin_amdgcn_readfirstlane(rec.z);
                const float dx = __int_as_float(rec.x);
                const float dy = __int_as_float(rec.y);
                const float* sp = dataB + soff;   // uniform base (saddr form)

                v2f bA, bB;
                bA.x = sp[vIdxA];
                bA.y = sp[vIdxA + 1];
                bB.x = sp[vIdxB];
                bB.y = sp[vIdxB + 1];

                const float e = fmaf(dy, hsel, hbase);
                v2f a;
                a.x = (1.0f - dx) * e;
                a.y = dx * e;

                accA = __builtin_amdgcn_wmma_f32_16x16x4_f32(
                    false, a, false, bA, (short)0, accA, false, false);
                accB = __builtin_amdgcn_wmma_f32_16x16x4_f32(
                    false, a, false, bB, (short)0, accB, false, false);
            }
        }
        // D VGPR0: lanes 0-15 = (M=0, N=lane), lanes 16-31 = (M=8, N=lane-16);
        // all M rows equal -> lane L's channel-in-block is wave*32 + L.
        const float res = (half ? accB[0] : accA[0]) * s_inv[bin];
        s_res[(wave * 32 + lane) * NBIN + bin] = res;
    }
    __syncthreads();

    // ---- phase 3: coalesced writeout of this ROI's 128-channel slab ----
    float* dst = out + ((size_t)r * C_FEAT + (size_t)cg * CH_PER_BLK) * NBIN;
    for (int i = tid; i < CH_PER_BLK * NBIN; i += BLK_THREADS)
        dst[i] = s_res[i];
}

extern "C" void kernel_launch(void* const* d_in, const int* in_sizes, int n_in,
                              void* d_out, int out_size, void* d_ws, size_t ws_size,
                              hipStream_t stream) {
    const float* data   = (const float*)d_in[0];
    const float* rois   = (const float*)d_in[1];
    const float* offset = (const float*)d_in[2];
    float* out = (float*)d_out;

    const int n_rois = in_sizes[1] / 5;
    dim3 grid(n_rois, C_FEAT / CH_PER_BLK);
    deform_roi_pool_wmma<<<grid, BLK_THREADS, 0, stream>>>(data, rois, offset, out);
}